// T3A_7284264534066
// MI455X (gfx1250) — compile-verified
//
#include <hip/hip_runtime.h>
#include <hip/hip_bf16.h>
#include <math.h>

#define NUM_CLASSES 400
#define FILTER_K    100
#define FEAT        768
#define NSUP        200000
#define NB          64
#define NTOT        (NSUP + NB)
#define CLS_CAP     1024
#define SEL_CAP     128
#define EPSN        1e-12f

typedef __attribute__((ext_vector_type(2))) float v2f;
typedef __attribute__((ext_vector_type(8))) float v8f;

// -------------------------------------------------------------------------
// K1/K5: C[64,400] = A[64,768] @ Bt[400,768]^T (+bias). One wave per 16x16
// C tile, V_WMMA_F32_16X16X4_F32 over K in steps of 4.
// ISA layouts (CDNA5 ISA):
//   A 16x4 : lanes 0-15 row M=lane (K=0,1 in v0,v1); lanes 16-31 same rows, K=2,3
//   B 4x16 : lanes 0-15 col N=lane (K=0,1); lanes 16-31 same cols, K=2,3
//   C/D    : VGPR r -> M=r (lanes 0-15) / M=r+8 (lanes 16-31), N=lane%16
// -------------------------------------------------------------------------
__global__ __launch_bounds__(32)
void ttaK_gemm_nt_wmma(const float* __restrict__ A, const float* __restrict__ Bt,
                       const float* __restrict__ bias, float* __restrict__ C,
                       int useBias) {
  const int lane = threadIdx.x;
  const int half = lane >> 4;        // 0: K pair {0,1}, 1: K pair {2,3}
  const int l    = lane & 15;
  const int m0 = blockIdx.x * 16;    // 4 tiles of M=64
  const int n0 = blockIdx.y * 16;    // 25 tiles of N=400
  const float* __restrict__ arow = A  + (size_t)(m0 + l) * FEAT;
  const float* __restrict__ brow = Bt + (size_t)(n0 + l) * FEAT;

  v8f acc = {};
#pragma unroll 4
  for (int k = 0; k < FEAT; k += 4) {
    const int ka = k + half * 2;
    v2f a = *(const v2f*)(arow + ka);
    v2f b = *(const v2f*)(brow + ka);
    acc = __builtin_amdgcn_wmma_f32_16x16x4_f32(
        /*neg_a=*/false, a, /*neg_b=*/false, b,
        /*c_mod=*/(short)0, acc, /*reuse_a=*/false, /*reuse_b=*/false);
  }

  const int n = n0 + l;
  const float bv = useBias ? bias[n] : 0.0f;
#pragma unroll
  for (int r = 0; r < 8; ++r) {
    const int m = m0 + r + half * 8;
    C[(size_t)m * NUM_CLASSES + n] = acc[r] + bv;
  }
}

// -------------------------------------------------------------------------
// K2: per batch row: softmax entropy + argmax (first max wins, like jnp.argmax)
// One wave (32 lanes) per row.
// -------------------------------------------------------------------------
__global__ __launch_bounds__(32)
void ttaK_row_stats(const float* __restrict__ P, float* __restrict__ ent_b,
                    int* __restrict__ yhat) {
  const int row = blockIdx.x;
  const int lane = threadIdx.x;
  const float* __restrict__ p = P + (size_t)row * NUM_CLASSES;

  float vmax = -3.402823466e38f;
  int imax = NUM_CLASSES;
  for (int c = lane; c < NUM_CLASSES; c += 32) {
    float v = p[c];
    if (v > vmax) { vmax = v; imax = c; }   // strictly-greater keeps first index
  }
#pragma unroll
  for (int off = 16; off > 0; off >>= 1) {
    float ov = __shfl_xor(vmax, off, 32);
    int   oi = __shfl_xor(imax, off, 32);
    if (ov > vmax || (ov == vmax && oi < imax)) { vmax = ov; imax = oi; }
  }

  float se = 0.0f, sw = 0.0f;   // sum exp(d), sum d*exp(d), d = p - max
  for (int c = lane; c < NUM_CLASSES; c += 32) {
    float d = p[c] - vmax;
    float e = __expf(d);
    se += e; sw += d * e;
  }
#pragma unroll
  for (int off = 16; off > 0; off >>= 1) {
    se += __shfl_xor(se, off, 32);
    sw += __shfl_xor(sw, off, 32);
  }
  if (lane == 0) {
    ent_b[row] = __logf(se) - sw / se;
    yhat[row]  = imax;
  }
}

// -------------------------------------------------------------------------
// K3: one block per class. Deterministically gather (ent, idx) of all members
// in ascending index order (ballot/prefix compaction), rank each by
// (ent, idx) lexicographic, keep rank < FILTER_K, compact kept indices in
// index order. Label/entropy arrays are L2-resident, so the 400x scan is cheap.
// -------------------------------------------------------------------------
__global__ __launch_bounds__(256)
void ttaK_select(const int* __restrict__ labels, const float* __restrict__ ent,
                 const int* __restrict__ yhat, const float* __restrict__ ent_b,
                 int* __restrict__ sel_idx, int* __restrict__ sel_count) {
  const int c = blockIdx.x;
  const int tid = threadIdx.x;
  const int wid = tid >> 5, lane = tid & 31;

  __shared__ float se[CLS_CAP];
  __shared__ int   si[CLS_CAP];
  __shared__ unsigned char keep[CLS_CAP];
  __shared__ int warp_cnt[8];
  __shared__ int s_n;
  if (tid == 0) s_n = 0;
  __syncthreads();

  for (int base = 0; base < NTOT; base += 256) {
    const int i = base + tid;
    int y = -1; float e = 0.0f;
    if (i < NTOT) {
      if (i < NSUP) { y = labels[i]; e = ent[i]; }
      else          { y = yhat[i - NSUP]; e = ent_b[i - NSUP]; }
    }
    const bool take = (y == c);
    const unsigned m = (unsigned)__ballot(take);   // wave32: low 32 bits
    if (lane == 0) warp_cnt[wid] = __popc(m);
    const int wprefix = __popc(m & ((1u << lane) - 1u));
    __syncthreads();
    const int chunk_base = s_n;
    int wbase = 0, total = 0;
#pragma unroll
    for (int w = 0; w < 8; ++w) {
      if (w < wid) wbase += warp_cnt[w];
      total += warp_cnt[w];
    }
    const int pos = chunk_base + wbase + wprefix;
    if (take && pos < CLS_CAP) { se[pos] = e; si[pos] = i; }
    __syncthreads();
    if (tid == 0) s_n = min(chunk_base + total, CLS_CAP);
    __syncthreads();
  }

  const int n = s_n;
  for (int t = tid; t < n; t += 256) {
    const float et = se[t];
    const int   it = si[t];
    int rank = 0;
    for (int j = 0; j < n; ++j) {
      const float ej = se[j];
      rank += (ej < et) || (ej == et && si[j] < it);
    }
    keep[t] = (rank < FILTER_K) ? 1 : 0;
  }
  __syncthreads();
  if (tid == 0) {                      // n <= 1024, trivial serial compact
    int cnt = 0;
    for (int t = 0; t < n && cnt < SEL_CAP; ++t)
      if (keep[t]) sel_idx[c * SEL_CAP + cnt++] = si[t];
    sel_count[c] = cnt;
  }
}

// -------------------------------------------------------------------------
// K4: one block (256 threads) per class. For each selected row: load 768
// floats (3/thread, coalesced b32), block-reduce sum of squares, accumulate
// row/max(||row||,eps). Then normalize the class sum and write Wn^T[c].
// Gathers only kept rows: ~123 MB total HBM traffic -> the roofline floor.
// -------------------------------------------------------------------------
__global__ __launch_bounds__(256)
void ttaK_accum(const float* __restrict__ supports, const float* __restrict__ x,
                const int* __restrict__ sel_idx, const int* __restrict__ sel_count,
                float* __restrict__ Wnt) {
  const int c = blockIdx.x;
  const int tid = threadIdx.x;
  __shared__ float red[256];
  __shared__ float s_scale;

  float a0 = 0.0f, a1 = 0.0f, a2 = 0.0f;
  const int cnt = sel_count[c];

  for (int s = 0; s < cnt; ++s) {
    const int idx = sel_idx[c * SEL_CAP + s];
    const float* __restrict__ row = (idx < NSUP)
        ? (supports + (size_t)idx * FEAT)
        : (x + (size_t)(idx - NSUP) * FEAT);
    const float v0 = row[tid];
    const float v1 = row[tid + 256];
    const float v2 = row[tid + 512];
    red[tid] = v0 * v0 + v1 * v1 + v2 * v2;
    __syncthreads();
#pragma unroll
    for (int off = 128; off > 0; off >>= 1) {
      if (tid < off) red[tid] += red[tid + off];
      __syncthreads();
    }
    if (tid == 0) s_scale = 1.0f / fmaxf(sqrtf(red[0]), EPSN);
    __syncthreads();
    const float sc = s_scale;
    a0 += v0 * sc; a1 += v1 * sc; a2 += v2 * sc;
  }

  red[tid] = a0 * a0 + a1 * a1 + a2 * a2;
  __syncthreads();
#pragma unroll
  for (int off = 128; off > 0; off >>= 1) {
    if (tid < off) red[tid] += red[tid + off];
    __syncthreads();
  }
  if (tid == 0) s_scale = 1.0f / fmaxf(sqrtf(red[0]), EPSN);
  __syncthreads();
  const float sc = s_scale;
  float* __restrict__ w = Wnt + (size_t)c * FEAT;
  w[tid]       = a0 * sc;
  w[tid + 256] = a1 * sc;
  w[tid + 512] = a2 * sc;
}

// -------------------------------------------------------------------------
// Host-side orchestration. Workspace layout (floats):
//   [0, 25600)          P          64x400 logits
//   [25600, 25664)      ent_b      64
//   [25664, 25728)      yhat       64 (int)
//   [25728, 26128)      sel_count  400 (int)
//   [26128, 77328)      sel_idx    400x128 (int)
//   [77328, 384528)     Wn^T       400x768
// ~1.5 MB total.
// -------------------------------------------------------------------------
extern "C" void kernel_launch(void* const* d_in, const int* in_sizes, int n_in,
                              void* d_out, int out_size, void* d_ws, size_t ws_size,
                              hipStream_t stream) {
  const float* x        = (const float*)d_in[0];
  const float* Wc       = (const float*)d_in[1];
  const float* bias     = (const float*)d_in[2];
  const float* supports = (const float*)d_in[3];
  const int*   labels   = (const int*)d_in[4];
  const float* ent      = (const float*)d_in[5];
  float* out = (float*)d_out;

  float* ws        = (float*)d_ws;
  float* P         = ws;
  float* ent_b     = ws + 25600;
  int*   yhat      = (int*)(ws + 25664);
  int*   sel_count = (int*)(ws + 25728);
  int*   sel_idx   = (int*)(ws + 26128);
  float* Wnt       = ws + 77328;

  dim3 gemm_grid(NB / 16, NUM_CLASSES / 16);   // 4 x 25 tiles

  ttaK_gemm_nt_wmma<<<gemm_grid, 32, 0, stream>>>(x, Wc, bias, P, 1);
  ttaK_row_stats<<<NB, 32, 0, stream>>>(P, ent_b, yhat);
  ttaK_select<<<NUM_CLASSES, 256, 0, stream>>>(labels, ent, yhat, ent_b,
                                               sel_idx, sel_count);
  ttaK_accum<<<NUM_CLASSES, 256, 0, stream>>>(supports, x, sel_idx, sel_count, Wnt);
  ttaK_gemm_nt_wmma<<<gemm_grid, 32, 0, stream>>>(x, Wnt, nullptr, out, 0);
}